// CapsNet_82617990906075
// MI455X (gfx1250) — compile-verified
//
#include <hip/hip_runtime.h>

typedef __attribute__((ext_vector_type(2))) float v2f;
typedef __attribute__((ext_vector_type(8))) float v8f;

// ---------------- LDS layout (floats, dynamic shared) ----------------
// x tile   : [16][210]
// u        : [16][7][8]
// u_hat    : [16][UH_STRIDE]   ( [n][m][k] = n*192 + m*16 + k, padded )
// b logits : [16][84]
// v        : [16][V_STRIDE]    ( m*16 + k, padded )
constexpr int X_OFF     = 0;
constexpr int U_OFF     = X_OFF + 16 * 210;          // 3360
constexpr int UH_OFF    = U_OFF + 16 * 56;           // +896
constexpr int UH_STRIDE = 1364;                      // 7*12*16=1344, padded (mod 8 != 0)
constexpr int B_OFF     = UH_OFF + 16 * UH_STRIDE;
constexpr int V_OFF     = B_OFF + 16 * 84;
constexpr int V_STRIDE  = 200;                       // 12*16=192, padded
constexpr int LDS_FLOATS = V_OFF + 16 * V_STRIDE;
constexpr int LDS_BYTES  = LDS_FLOATS * 4;           // ~122.5 KB

constexpr float EPSV = 1e-8f;

__global__ __launch_bounds__(128, 1)
void capsnet_wmma_kernel(const float* __restrict__ x,
                         const float* __restrict__ W_pc,
                         const float* __restrict__ b_pc,
                         const float* __restrict__ W,
                         const int*   __restrict__ d_niter,
                         float*       __restrict__ out)
{
    extern __shared__ float lds[];
    const int tid   = threadIdx.x;
    const int lane  = tid & 31;
    const int wave  = tid >> 5;
    const int gbase = blockIdx.x * 16;               // 16 samples per block
    const int niter = d_niter[0];

    // ---- stage x tile into LDS (fully coalesced, x is read exactly once) ----
    for (int i = tid; i < 16 * 210; i += 128)
        lds[X_OFF + i] = x[(size_t)gbase * 210 + i];
    // zero routing logits
    for (int i = tid; i < 16 * 84; i += 128)
        lds[B_OFF + i] = 0.0f;
    __syncthreads();

    // ---- step A: primary capsules u = squash(groups @ W_pc^T + b_pc) ----
    {
        const int s  = tid >> 3;                     // sample 0..15
        const int og = tid & 7;                      // capsule dim 0..7
        const float* xs = &lds[X_OFF + s * 210];
        for (int n = 0; n < 7; ++n) {
            float acc = b_pc[n * 8 + og];
            const float* wp = &W_pc[(n * 8 + og) * 30];
            const float* xg = xs + n * 30;
            #pragma unroll
            for (int g = 0; g < 30; ++g) acc += xg[g] * wp[g];
            // ||u||^2 across the 8 capsule dims (8-lane group, wave32 shuffles)
            float p = acc * acc;
            p += __shfl_xor(p, 1);
            p += __shfl_xor(p, 2);
            p += __shfl_xor(p, 4);
            const float norm = sqrtf(p);
            const float f = norm / ((1.0f + p) * (norm + EPSV)); // squash scale
            lds[U_OFF + (s * 7 + n) * 8 + og] = acc * f;
        }
    }
    __syncthreads();

    // ---- step B: u_hat[b,n,m,k] = sum_j u[b,n,j] * W[n,m,j,k]  (fp32 WMMA) ----
    // GEMM per n: M=16 samples, K=8 (2 k-steps of 4), N=192 (12 tiles of 16).
    {
        const int M    = lane & 15;                  // A: row / B: column / D: k-dim
        const int half = lane >> 4;                  // lane half selects K pair
        for (int n = wave; n < 7; n += 4) {          // uniform per wave -> EXEC all-1s
            for (int mc = 0; mc < 12; ++mc) {
                v8f acc = {};
                #pragma unroll
                for (int ks = 0; ks < 2; ++ks) {
                    const int kb = ks * 4 + half * 2;     // K index for this lane half
                    v2f a, b;
                    // A (16x4 f32): lanes 0-15 -> K=kb,kb+1 ; lanes 16-31 -> K=kb,kb+1 (kb+2 base)
                    a.x = lds[U_OFF + (M * 7 + n) * 8 + kb];
                    a.y = lds[U_OFF + (M * 7 + n) * 8 + kb + 1];
                    // B (4x16 f32): row K striped per VGPR, N = lane&15
                    const float* wb = &W[((size_t)(n * 12 + mc) * 8 + kb) * 16 + M];
                    b.x = wb[0];
                    b.y = wb[16];
                    acc = __builtin_amdgcn_wmma_f32_16x16x4_f32(
                        false, a, false, b, (short)0, acc, false, false);
                }
                // D (16x16 f32): VGPR v -> sample v (+8 for upper half), lane&15 -> k-dim
                #pragma unroll
                for (int v = 0; v < 8; ++v) {
                    const int s = v + half * 8;
                    lds[UH_OFF + s * UH_STRIDE + (n * 12 + mc) * 16 + M] = acc[v];
                }
            }
        }
    }
    __syncthreads();   // u_hat for a sample was written by all 4 waves

    // ---- dynamic routing: thread <-> (sample, k-group); all state wave-private ----
    {
        const int s  = tid >> 3;                     // sample 0..15
        const int kg = tid & 7;                      // handles k = kg and kg+8
        const int k0 = kg, k1 = kg + 8;
        const float* uh = &lds[UH_OFF + s * UH_STRIDE];
        float* bl = &lds[B_OFF + s * 84];
        float* vl = &lds[V_OFF + s * V_STRIDE];

        for (int it = 0; it < niter; ++it) {
            float s0[12], s1[12];
            #pragma unroll
            for (int m = 0; m < 12; ++m) { s0[m] = 0.0f; s1[m] = 0.0f; }

            for (int n = 0; n < 7; ++n) {
                // softmax over the 12 output capsules
                float bv[12], mx = -3.4e38f;
                #pragma unroll
                for (int m = 0; m < 12; ++m) { bv[m] = bl[n * 12 + m]; mx = fmaxf(mx, bv[m]); }
                float sum = 0.0f;
                #pragma unroll
                for (int m = 0; m < 12; ++m) { bv[m] = __expf(bv[m] - mx); sum += bv[m]; }
                const float inv = 1.0f / sum;
                const float* uhn = uh + n * 192;
                #pragma unroll
                for (int m = 0; m < 12; ++m) {
                    const float c = bv[m] * inv;
                    s0[m] += c * uhn[m * 16 + k0];
                    s1[m] += c * uhn[m * 16 + k1];
                }
            }

            const bool last = (it == niter - 1);
            #pragma unroll
            for (int m = 0; m < 12; ++m) {
                float p = s0[m] * s0[m] + s1[m] * s1[m];
                p += __shfl_xor(p, 1);
                p += __shfl_xor(p, 2);
                p += __shfl_xor(p, 4);
                const float norm = sqrtf(p);
                const float f = norm / ((1.0f + p) * (norm + EPSV));
                if (last) {
                    // ||v|| = ||s|| * f = p / ((1+p)(norm+eps))
                    if (kg == 0)
                        out[(size_t)(gbase + s) * 12 + m] = p / ((1.0f + p) * (norm + EPSV));
                } else {
                    vl[m * 16 + k0] = s0[m] * f;
                    vl[m * 16 + k1] = s1[m] * f;
                }
            }

            if (!last) {
                // b[n,m] += <u_hat[n,m,:], v[m,:]> ; each (n,m) owned by one thread
                for (int pidx = kg; pidx < 84; pidx += 8) {
                    const int n = pidx / 12, m = pidx % 12;
                    const float* uv = uh + n * 192 + m * 16;
                    const float* vv = vl + m * 16;
                    float dot = 0.0f;
                    #pragma unroll
                    for (int k = 0; k < 16; ++k) dot += uv[k] * vv[k];
                    bl[pidx] += dot;
                }
            }
        }
    }
}

extern "C" void kernel_launch(void* const* d_in, const int* in_sizes, int n_in,
                              void* d_out, int out_size, void* d_ws, size_t ws_size,
                              hipStream_t stream) {
    const float* x     = (const float*)d_in[0];
    const float* W_pc  = (const float*)d_in[1];
    const float* b_pc  = (const float*)d_in[2];
    const float* W     = (const float*)d_in[3];
    const int*   niter = (const int*)d_in[4];
    float* out = (float*)d_out;

    const int B      = in_sizes[0] / 210;            // 65536
    const int blocks = B / 16;                       // 16 samples per block

    hipFuncSetAttribute(reinterpret_cast<const void*>(capsnet_wmma_kernel),
                        hipFuncAttributeMaxDynamicSharedMemorySize, LDS_BYTES);
    capsnet_wmma_kernel<<<blocks, 128, LDS_BYTES, stream>>>(x, W_pc, b_pc, W, niter, out);
}